// SparseNodeTransformerLayer_90443421319561
// MI455X (gfx1250) — compile-verified
//
#include <hip/hip_runtime.h>
#include <math.h>

// SparseNodeTransformerLayer for MI455X (gfx1250, wave32).
// fp32 end-to-end; GEMMs via V_WMMA_F32_16X16X4_F32 (GEMM FLOPs are negligible
// vs. the L2-resident edge phase, so full fp32 precision is free).
// A-tile staging uses GLOBAL_LOAD_ASYNC_TO_LDS_B128 (ASYNCcnt path).

typedef __attribute__((ext_vector_type(2))) float v2f;
typedef __attribute__((ext_vector_type(8))) float v8f;

#define HID 128
#define NH  8
#define HD  16

// ---------------------------------------------------------------- zero init
__global__ void zero_f32_kernel(float* __restrict__ p, int n) {
  int i = blockIdx.x * blockDim.x + threadIdx.x;
  if (i < n) p[i] = 0.0f;
}

// ------------------------------------------------- async 8KB tile -> LDS
// Copies a contiguous 16x128-float (8 KB) tile from global into LDS using the
// CDNA5 async global->LDS path. Each of the 256 threads moves 2x16 bytes; the
// ISA adds INST_OFFSET to both the global and LDS addresses, so one address
// pair + offset:4096 covers both halves. Low 32 bits of a generic pointer to
// LDS are the wave-relative LDS byte address (ISA 10.2 aperture rules).
__device__ __forceinline__ void async_stage_tile(const float* __restrict__ gsrc,
                                                 float* lds) {
  const int t = threadIdx.x;
  const unsigned lds_addr =
      (unsigned)(size_t)(void*)lds + (unsigned)(t * 16);
  const unsigned long long gaddr =
      (unsigned long long)(size_t)gsrc + (unsigned long long)(t * 16);
  asm volatile(
      "global_load_async_to_lds_b128 %0, %1, off\n\t"
      "global_load_async_to_lds_b128 %0, %1, off offset:4096\n\t"
      "s_wait_asynccnt 0x0"
      :
      : "v"(lds_addr), "v"(gaddr)
      : "memory");
  __syncthreads();
}

// ---------------------------------------------------------------- WMMA tile
// Computes one 16x16 output tile of  out[n, j] = sum_k A[n,k] * W[j,k] + b[j]
// A tile (16 x 128) is in LDS; W is row-major [128,128] in global (L2-hot).
// Layout per CDNA5 ISA 7.12.2:
//   A (16x4 f32): lanes 0-15 -> K=0,1 in v[0],v[1]; lanes 16-31 -> K=2,3
//   B (4x16 f32): v[0] = row K=(0|2), v[1] = row K=(1|3), N = lane&15
//   C/D: VGPR r -> M = r + 8*(lane>=16), N = lane&15
__device__ __forceinline__ v8f wmma_tile_f32(const float* __restrict__ Atile,
                                             const float* __restrict__ W,
                                             int col, int half, int l16) {
  v8f acc = {};
#pragma unroll
  for (int kk = 0; kk < HID; kk += 4) {
    const int k0 = kk + half * 2;
    v2f a;
    a[0] = Atile[l16 * HID + k0];
    a[1] = Atile[l16 * HID + k0 + 1];
    const float2 wb = *(const float2*)(W + (size_t)col * HID + k0);
    v2f b;
    b[0] = wb.x;
    b[1] = wb.y;
    // 8 args: (neg_a, A, neg_b, B, c_mod, C, reuse_a, reuse_b)
    acc = __builtin_amdgcn_wmma_f32_16x16x4_f32(false, a, false, b, (short)0,
                                                acc, false, false);
  }
  return acc;
}

// ------------------------------------------------------- fused QKV projection
__global__ __launch_bounds__(256) void qkv_proj_kernel(
    const float* __restrict__ h,
    const float* __restrict__ Wq, const float* __restrict__ bq,
    const float* __restrict__ Wk, const float* __restrict__ bk,
    const float* __restrict__ Wv, const float* __restrict__ bv,
    float* __restrict__ q, float* __restrict__ k, float* __restrict__ v) {
  __shared__ __align__(16) float Atile[16 * HID];
  const int nodeBase = blockIdx.x * 16;

  async_stage_tile(h + (size_t)nodeBase * HID, Atile);

  const int lane = threadIdx.x & 31;
  const int wave = threadIdx.x >> 5;  // 0..7 -> output column tile
  const int half = lane >> 4;
  const int l16  = lane & 15;
  const int col  = wave * 16 + l16;

  const float* Ws[3]   = {Wq, Wk, Wv};
  const float* bias[3] = {bq, bk, bv};
  float* outs[3]       = {q, k, v};

#pragma unroll
  for (int m = 0; m < 3; ++m) {
    v8f acc = wmma_tile_f32(Atile, Ws[m], col, half, l16);
    const float b = bias[m][col];
    float* O = outs[m];
#pragma unroll
    for (int r = 0; r < 8; ++r) {
      const int row = half * 8 + r;
      O[(size_t)(nodeBase + row) * HID + col] = acc[r] + b;
    }
  }
}

// ------------------------------------------- edge scores + exp + denominator
// One thread per (edge, head). q/k are L2-resident (25.6 MB each).
__global__ __launch_bounds__(256) void edge_scores_kernel(
    const float* __restrict__ q, const float* __restrict__ k,
    const int* __restrict__ src, const int* __restrict__ dst,
    float* __restrict__ attn, float* __restrict__ z, int EH) {
  const int t = blockIdx.x * blockDim.x + threadIdx.x;
  if (t >= EH) return;
  const int e  = t >> 3;
  const int hh = t & 7;
  const int s = src[e];
  const int d = dst[e];
  const float4* qv = (const float4*)(q + (size_t)d * HID + hh * HD);
  const float4* kv = (const float4*)(k + (size_t)s * HID + hh * HD);
  float dot = 0.0f;
#pragma unroll
  for (int i = 0; i < 4; ++i) {
    const float4 a = qv[i];
    const float4 b = kv[i];
    dot += a.x * b.x + a.y * b.y + a.z * b.z + a.w * b.w;
  }
  float sc = dot * 0.25f;  // 1/sqrt(D), D=16
  sc = fminf(5.0f, fmaxf(-5.0f, sc));
  // Scores are clamped to [-5,5] -> exp never overflows, so the softmax
  // max-subtraction can be dropped exactly (shift invariance).
  const float ex = __expf(sc);
  attn[(size_t)e * NH + hh] = ex;
  atomicAdd(&z[(size_t)d * NH + hh], ex);
}

// ------------------------------------------ normalize + weighted scatter-add
__global__ __launch_bounds__(256) void edge_aggregate_kernel(
    const float* __restrict__ v, const int* __restrict__ src,
    const int* __restrict__ dst, const float* __restrict__ attn,
    const float* __restrict__ z, float* __restrict__ agg, int EH) {
  const int t = blockIdx.x * blockDim.x + threadIdx.x;
  if (t >= EH) return;
  const int e  = t >> 3;
  const int hh = t & 7;
  const int s = src[e];
  const int d = dst[e];
  const float w = attn[(size_t)e * NH + hh] / z[(size_t)d * NH + hh];
  const float4* vv = (const float4*)(v + (size_t)s * HID + hh * HD);
  float* o = agg + (size_t)d * HID + hh * HD;
#pragma unroll
  for (int i = 0; i < 4; ++i) {
    const float4 t4 = vv[i];
    atomicAdd(o + 4 * i + 0, w * t4.x);
    atomicAdd(o + 4 * i + 1, w * t4.y);
    atomicAdd(o + 4 * i + 2, w * t4.z);
    atomicAdd(o + 4 * i + 3, w * t4.w);
  }
}

// -------------------------------------------------------- output projection
__global__ __launch_bounds__(256) void out_proj_kernel(
    const float* __restrict__ x, const float* __restrict__ Wo,
    const float* __restrict__ bo, float* __restrict__ out) {
  __shared__ __align__(16) float Atile[16 * HID];
  const int nodeBase = blockIdx.x * 16;

  async_stage_tile(x + (size_t)nodeBase * HID, Atile);

  const int lane = threadIdx.x & 31;
  const int wave = threadIdx.x >> 5;
  const int half = lane >> 4;
  const int l16  = lane & 15;
  const int col  = wave * 16 + l16;

  v8f acc = wmma_tile_f32(Atile, Wo, col, half, l16);
  const float b = bo[col];
#pragma unroll
  for (int r = 0; r < 8; ++r) {
    const int row = half * 8 + r;
    out[(size_t)(nodeBase + row) * HID + col] = acc[r] + b;
  }
}

// ------------------------------------------------------------------- launch
extern "C" void kernel_launch(void* const* d_in, const int* in_sizes, int n_in,
                              void* d_out, int out_size, void* d_ws,
                              size_t ws_size, hipStream_t stream) {
  const float* h  = (const float*)d_in[0];
  const int* src  = (const int*)d_in[1];
  const int* dst  = (const int*)d_in[2];
  const float* Wq = (const float*)d_in[3];
  const float* bq = (const float*)d_in[4];
  const float* Wk = (const float*)d_in[5];
  const float* bk = (const float*)d_in[6];
  const float* Wv = (const float*)d_in[7];
  const float* bv = (const float*)d_in[8];
  const float* Wo = (const float*)d_in[9];
  const float* bo = (const float*)d_in[10];
  float* out = (float*)d_out;

  const int N = in_sizes[0] / HID;  // 50000
  const int E = in_sizes[1];        // 800000

  // Workspace layout (floats): q | k | v | agg | z | attn  (~130 MB)
  float* ws   = (float*)d_ws;
  float* q    = ws;
  float* k    = q + (size_t)N * HID;
  float* v    = k + (size_t)N * HID;
  float* agg  = v + (size_t)N * HID;
  float* z    = agg + (size_t)N * HID;  // agg and z contiguous
  float* attn = z + (size_t)N * NH;

  // Zero agg + z in one pass (they are adjacent).
  const int zn = N * (HID + NH);
  zero_f32_kernel<<<(zn + 255) / 256, 256, 0, stream>>>(agg, zn);

  // Q/K/V projections (WMMA).
  qkv_proj_kernel<<<N / 16, 256, 0, stream>>>(h, Wq, bq, Wk, bk, Wv, bv, q, k, v);

  // Edge phase.
  const int EH = E * NH;
  edge_scores_kernel<<<(EH + 255) / 256, 256, 0, stream>>>(q, k, src, dst, attn, z, EH);
  edge_aggregate_kernel<<<(EH + 255) / 256, 256, 0, stream>>>(v, src, dst, attn, z, agg, EH);

  // Output projection (WMMA).
  out_proj_kernel<<<N / 16, 256, 0, stream>>>(agg, Wo, bo, out);
}